// H2GCN_81527069213104
// MI455X (gfx1250) — compile-verified
//
#include <hip/hip_runtime.h>

// H2GCN pipeline for MI455X (gfx1250, wave32).
// Dense GEMMs via V_WMMA_F32_16X16X4_F32; SpMM via edge-parallel
// global_atomic_add_f32 (output rows live in the 192MB L2).

#define N_NODES 50000
#define IN_C    512
#define HID     128
#define OUT_C   64
#define NE1     800000
#define NE2     3200000
#define BN_EPS  1e-5f

typedef __attribute__((ext_vector_type(2))) float v2f;
typedef __attribute__((ext_vector_type(8))) float v8f;

// ---------------------------------------------------------------- zero fill
__global__ void __launch_bounds__(256) k_zero(float4* __restrict__ p, size_t n4) {
    float4 z = make_float4(0.f, 0.f, 0.f, 0.f);
    size_t i = (size_t)blockIdx.x * blockDim.x + threadIdx.x;
    size_t stride = (size_t)gridDim.x * blockDim.x;
    for (; i < n4; i += stride) p[i] = z;
}

// ------------------------------------------------------- WMMA fp32 K-segment
// Accumulates a 16x16 (rows x cols) tile of A[rowBase: , :] @ W[wRowBase: , colBase:]
// A: row-major [*, aStride], W: row-major [*, wStride]. kLen multiple of 4.
__device__ __forceinline__ v8f wmma_f32_seg(v8f acc,
        const float* __restrict__ A, int aStride,
        const float* __restrict__ W, int wStride, int wRowBase, int kLen,
        int rowBase, int colBase, int half, int l16) {
    const float* ar = A + (size_t)(rowBase + l16) * aStride + 2 * half;
    for (int k = 0; k < kLen; k += 4) {
        v2f a, b;
        // A 16x4: lane half h, element j -> K = k + 2h + j, M = l16
        a[0] = ar[k + 0];
        a[1] = ar[k + 1];
        // B 4x16: lane half h, element j -> K = k + 2h + j, N = l16
        const float* wr = W + (size_t)(wRowBase + k + 2 * half) * wStride + colBase + l16;
        b[0] = wr[0];
        b[1] = wr[wStride];
        acc = __builtin_amdgcn_wmma_f32_16x16x4_f32(
            /*neg_a=*/false, a, /*neg_b=*/false, b,
            /*c_mod=*/(short)0, acc, /*reuse_a=*/false, /*reuse_b=*/false);
    }
    return acc;
}

// -------------------------------------------------- h0 = relu(x@We + be)
// grid = (N/16, HID/16), block = 32 (one wave per 16x16 output tile)
__global__ void __launch_bounds__(32) k_embed(
        const float* __restrict__ x, const float* __restrict__ w,
        const float* __restrict__ bias, float* __restrict__ h0) {
    const int rowBase = blockIdx.x * 16;
    const int colBase = blockIdx.y * 16;
    const int lane = threadIdx.x;
    const int half = lane >> 4;
    const int l16  = lane & 15;
    v8f acc = {};
    acc = wmma_f32_seg(acc, x, IN_C, w, HID, 0, IN_C, rowBase, colBase, half, l16);
    const float bv = bias[colBase + l16];
    #pragma unroll
    for (int i = 0; i < 8; ++i) {
        // C/D layout: VGPR i -> M = i + 8*half, N = l16
        const int row = rowBase + half * 8 + i;
        float v = acc[i] + bv;
        h0[(size_t)row * HID + colBase + l16] = fmaxf(v, 0.0f);
    }
}

// ------------------------------------------------ SpMM: out[dst] += w*feat[src]
// One edge per 2^cshift consecutive threads; each thread moves a float4.
// Lanes of a wave cover one edge's whole feature row -> coalesced gather and
// coalesced atomics into the (L2-resident) destination row.
__global__ void __launch_bounds__(256) k_spmm(
        const int* __restrict__ src, const int* __restrict__ dst,
        const float* __restrict__ ew, const float* __restrict__ feat,
        int featStride, float* __restrict__ outp, int outStride, int colOff,
        unsigned nEdges, int cshift) {
    const unsigned tid = blockIdx.x * blockDim.x + threadIdx.x;
    const unsigned total = nEdges << cshift;
    if (tid >= total) return;
    const unsigned e = tid >> cshift;
    const int c = (int)((tid & ((1u << cshift) - 1u)) << 2);
    const int s = src[e];
    const int d = dst[e];
    const float w = ew[e];
    const float4 v = *(const float4*)(feat + (size_t)s * featStride + c);
    float* o = outp + (size_t)d * outStride + colOff + c;
    __hip_atomic_fetch_add(o + 0, w * v.x, __ATOMIC_RELAXED, __HIP_MEMORY_SCOPE_AGENT);
    __hip_atomic_fetch_add(o + 1, w * v.y, __ATOMIC_RELAXED, __HIP_MEMORY_SCOPE_AGENT);
    __hip_atomic_fetch_add(o + 2, w * v.z, __ATOMIC_RELAXED, __HIP_MEMORY_SCOPE_AGENT);
    __hip_atomic_fetch_add(o + 3, w * v.w, __ATOMIC_RELAXED, __HIP_MEMORY_SCOPE_AGENT);
}

// ---------------------------------------------------- BN stats (256 columns)
// block = 256 threads (thread t owns column t -> fully coalesced row reads),
// each block strides a chunk of rows, then one atomic per (block, column).
__global__ void __launch_bounds__(256) k_bn_stats(
        const float* __restrict__ h1, float* __restrict__ sums,
        float* __restrict__ sumsq, int n) {
    const int col = threadIdx.x;                 // 0..255
    const int rowsPerBlock = (n + gridDim.x - 1) / gridDim.x;
    const int r0 = blockIdx.x * rowsPerBlock;
    const int r1 = min(n, r0 + rowsPerBlock);
    float s = 0.f, q = 0.f;
    for (int r = r0; r < r1; ++r) {
        const float v = h1[(size_t)r * (2 * HID) + col];
        s += v;
        q += v * v;
    }
    __hip_atomic_fetch_add(&sums[col],  s, __ATOMIC_RELAXED, __HIP_MEMORY_SCOPE_AGENT);
    __hip_atomic_fetch_add(&sumsq[col], q, __ATOMIC_RELAXED, __HIP_MEMORY_SCOPE_AGENT);
}

__global__ void __launch_bounds__(256) k_bn_apply(
        float* __restrict__ h1, const float* __restrict__ sums,
        const float* __restrict__ sumsq, const float* __restrict__ gamma,
        const float* __restrict__ beta, int n) {
    const size_t i = (size_t)blockIdx.x * blockDim.x + threadIdx.x;
    if (i >= (size_t)n * (2 * HID)) return;
    const int col = (int)(i & (2 * HID - 1));
    const float invn = 1.0f / (float)n;
    const float mean = sums[col] * invn;
    const float var  = sumsq[col] * invn - mean * mean;
    const float inv  = rsqrtf(var + BN_EPS);
    h1[i] = gamma[col] * (h1[i] - mean) * inv + beta[col];
}

// -------------------------------------- out = [h0|h1|h2] @ w_last + b_last
// grid = (N/16, OUT_C/16), block = 32. K=896 walked as three segments so the
// jk concat is never materialized.
__global__ void __launch_bounds__(32) k_final(
        const float* __restrict__ h0, const float* __restrict__ h1,
        const float* __restrict__ h2, const float* __restrict__ w,
        const float* __restrict__ bias, float* __restrict__ outp) {
    const int rowBase = blockIdx.x * 16;
    const int colBase = blockIdx.y * 16;
    const int lane = threadIdx.x;
    const int half = lane >> 4;
    const int l16  = lane & 15;
    v8f acc = {};
    acc = wmma_f32_seg(acc, h0, HID,     w, OUT_C, 0,       HID,     rowBase, colBase, half, l16);
    acc = wmma_f32_seg(acc, h1, 2 * HID, w, OUT_C, HID,     2 * HID, rowBase, colBase, half, l16);
    acc = wmma_f32_seg(acc, h2, 4 * HID, w, OUT_C, 3 * HID, 4 * HID, rowBase, colBase, half, l16);
    const float bv = bias[colBase + l16];
    #pragma unroll
    for (int i = 0; i < 8; ++i) {
        const int row = rowBase + half * 8 + i;
        outp[(size_t)row * OUT_C + colBase + l16] = acc[i] + bv;
    }
}

// ------------------------------------------------------------------- launch
extern "C" void kernel_launch(void* const* d_in, const int* in_sizes, int n_in,
                              void* d_out, int out_size, void* d_ws, size_t ws_size,
                              hipStream_t stream) {
    const float* x        = (const float*)d_in[0];
    const float* w_embed  = (const float*)d_in[1];
    const float* b_embed  = (const float*)d_in[2];
    const float* bn_gamma = (const float*)d_in[3];
    const float* bn_beta  = (const float*)d_in[4];
    const float* w_last   = (const float*)d_in[5];
    const float* b_last   = (const float*)d_in[6];
    const float* ew1      = (const float*)d_in[7];
    const float* ew2      = (const float*)d_in[8];
    const int*   src1     = (const int*)d_in[9];
    const int*   dst1     = (const int*)d_in[10];
    const int*   src2     = (const int*)d_in[11];
    const int*   dst2     = (const int*)d_in[12];
    float* out = (float*)d_out;

    // workspace layout: [h0 | h1 | h2 | bn stats(512)]
    float* h0    = (float*)d_ws;
    float* h1    = h0 + (size_t)N_NODES * HID;
    float* h2    = h1 + (size_t)N_NODES * 2 * HID;
    float* stats = h2 + (size_t)N_NODES * 4 * HID;

    // zero h1, h2, stats (contiguous region)
    const size_t zf = (size_t)N_NODES * 2 * HID + (size_t)N_NODES * 4 * HID + 512;
    k_zero<<<2048, 256, 0, stream>>>((float4*)h1, zf / 4);

    // h0 = relu(x @ w_embed + b_embed)
    k_embed<<<dim3(N_NODES / 16, HID / 16), 32, 0, stream>>>(x, w_embed, b_embed, h0);

    // conv1: h1 = [A1@h0 | A2@h0], feat width 128 -> cshift 5
    {
        const unsigned t1 = (unsigned)NE1 << 5;
        const unsigned t2 = (unsigned)NE2 << 5;
        k_spmm<<<(t1 + 255) / 256, 256, 0, stream>>>(src1, dst1, ew1, h0, HID,
                                                     h1, 2 * HID, 0,   NE1, 5);
        k_spmm<<<(t2 + 255) / 256, 256, 0, stream>>>(src2, dst2, ew2, h0, HID,
                                                     h1, 2 * HID, HID, NE2, 5);
    }

    // batchnorm over h1 columns (in place)
    k_bn_stats<<<256, 256, 0, stream>>>(h1, stats, stats + 256, N_NODES);
    {
        const size_t tot = (size_t)N_NODES * 2 * HID;
        k_bn_apply<<<(unsigned)((tot + 255) / 256), 256, 0, stream>>>(
            h1, stats, stats + 256, bn_gamma, bn_beta, N_NODES);
    }

    // conv2: h2 = [A1@h1 | A2@h1], feat width 256 -> cshift 6
    {
        const unsigned t1 = (unsigned)NE1 << 6;
        const unsigned t2 = (unsigned)NE2 << 6;
        k_spmm<<<(t1 + 255) / 256, 256, 0, stream>>>(src1, dst1, ew1, h1, 2 * HID,
                                                     h2, 4 * HID, 0,       NE1, 6);
        k_spmm<<<(t2 + 255) / 256, 256, 0, stream>>>(src2, dst2, ew2, h1, 2 * HID,
                                                     h2, 4 * HID, 2 * HID, NE2, 6);
    }

    // out = [h0|h1|h2] @ w_last + b_last
    k_final<<<dim3(N_NODES / 16, OUT_C / 16), 32, 0, stream>>>(
        h0, h1, h2, w_last, b_last, out);

    (void)in_sizes; (void)n_in; (void)out_size; (void)ws_size;
}